// ContrastMemory_75917841924732
// MI455X (gfx1250) — compile-verified
//
#include <hip/hip_runtime.h>
#include <hip/hip_bf16.h>
#include <math.h>

// Problem constants (from the reference)
#define BB   64
#define KK   128
#define KP1  129         // K+1
#define DD   128
#define NN   50000
#define INV_T 20.0f      // 1/T
#define MOM  0.5f
#define EPS_ 1e-6f

typedef __attribute__((ext_vector_type(2))) float v2f;
typedef __attribute__((ext_vector_type(8))) float v8f;

// ---------------------------------------------------------------------------
// Gather anchor rows + squared norms:  anc[b, :] = mem[y[b], :], na[b] = ||anc||^2
// ---------------------------------------------------------------------------
__global__ __launch_bounds__(128) void prep_anchor_kernel(
    const float* __restrict__ mem, const int* __restrict__ y,
    float* __restrict__ anc, float* __restrict__ na)
{
    __shared__ float red[128];
    const int b = blockIdx.x;
    const int t = threadIdx.x;
    const int row = y[b];
    const float v = mem[(size_t)row * DD + t];
    anc[b * DD + t] = v;
    red[t] = v * v;
    __syncthreads();
    for (int s = 64; s > 0; s >>= 1) { if (t < s) red[t] += red[t + s]; __syncthreads(); }
    if (t == 0) na[b] = red[0];
}

// ---------------------------------------------------------------------------
// Gather weight rows (+eps) + squared norms:
//   wp[a,k,:] = mem[idx[a,k], :] + eps, nw[a,k] = ||wp||^2
// ---------------------------------------------------------------------------
__global__ __launch_bounds__(128) void prep_weight_kernel(
    const float* __restrict__ mem, const int* __restrict__ idx,
    float* __restrict__ wp, float* __restrict__ nw)
{
    __shared__ float red[128];
    const int blk = blockIdx.x;            // 0 .. BB*KP1-1
    const int t = threadIdx.x;
    const int row = idx[blk];
    const float v = mem[(size_t)row * DD + t] + EPS_;
    wp[(size_t)blk * DD + t] = v;
    red[t] = v * v;
    __syncthreads();
    for (int s = 64; s > 0; s >>= 1) { if (t < s) red[t] += red[t + s]; __syncthreads(); }
    if (t == 0) nw[blk] = red[0];
}

// ---------------------------------------------------------------------------
// Anchor relation: arn[a,b,:] = normalize(v[a] - anc[b] + eps)
//                  sab[a,b]   = dot(anc[b], arn[a,b])
// ---------------------------------------------------------------------------
__global__ __launch_bounds__(128) void prep_ar_kernel(
    const float* __restrict__ v, const float* __restrict__ anc,
    float* __restrict__ arn, float* __restrict__ sab)
{
    __shared__ float red[128];
    const int blk = blockIdx.x;            // 0 .. BB*BB-1
    const int a = blk >> 6;
    const int b = blk & 63;
    const int t = threadIdx.x;
    const float av = anc[b * DD + t];
    const float x = v[a * DD + t] - av + EPS_;
    red[t] = x * x;
    __syncthreads();
    for (int s = 64; s > 0; s >>= 1) { if (t < s) red[t] += red[t + s]; __syncthreads(); }
    float den = sqrtf(red[0]);
    __syncthreads();
    den = den < 1e-12f ? 1e-12f : den;
    const float xn = x / den;
    arn[((size_t)a * BB + b) * DD + t] = xn;
    red[t] = av * xn;
    __syncthreads();
    for (int s = 64; s > 0; s >>= 1) { if (t < s) red[t] += red[t + s]; __syncthreads(); }
    if (t == 0) sab[a * BB + b] = red[0];
}

// ---------------------------------------------------------------------------
// Fused dual-GEMM einsum via V_WMMA_F32_16X16X4_F32, one wave = one 16x16 tile.
//   C1[k,b] = dot(wp[a,k,:], arn[a,b,:])   (numerator GEMM)
//   C2[k,b] = dot(wp[a,k,:], anc[b,:])     (for ||diff||^2)
// Epilogue writes e = exp(clamp(cos,-1,1) * (1/T)) directly to eout.
// grid = (4 Ntiles, 9 Mtiles, 64 a), block = 32 (one wave; EXEC all ones).
// ---------------------------------------------------------------------------
__global__ __launch_bounds__(32) void einsum_wmma_kernel(
    const float* __restrict__ wp,   // [BB, KP1, DD]
    const float* __restrict__ arn,  // [BB, BB, DD]
    const float* __restrict__ anc,  // [BB, DD]
    const float* __restrict__ sab,  // [BB, BB]
    const float* __restrict__ nw,   // [BB, KP1]
    const float* __restrict__ na,   // [BB]
    float* __restrict__ eout)       // [BB, BB, KP1]  ((a*BB+b)*KP1 + k)
{
    const int lane  = threadIdx.x;
    const int ntile = blockIdx.x;       // b tiles: 0..3
    const int mtile = blockIdx.y;       // k tiles: 0..8
    const int a     = blockIdx.z;       // 0..63

    const int half = lane >> 4;         // lanes 16-31 hold K=2,3 of each step
    const int l15  = lane & 15;

    int m = mtile * 16 + l15;           // A-row (k index); clamp for tile 8
    if (m > KK) m = KK;
    const float* Abase  = wp  + ((size_t)a * KP1 + m) * DD + 2 * half;
    const int    bcol   = ntile * 16 + l15;                // B-column (b index)
    const float* B1base = arn + ((size_t)a * BB + bcol) * DD + 2 * half;
    const float* B2base = anc + (size_t)bcol * DD + 2 * half;

    v8f c1 = {};
    v8f c2 = {};
#pragma unroll 4
    for (int kk = 0; kk < DD / 4; ++kk) {
        const v2f av = *(const v2f*)(Abase  + kk * 4);
        const v2f b1 = *(const v2f*)(B1base + kk * 4);
        const v2f b2 = *(const v2f*)(B2base + kk * 4);
        c1 = __builtin_amdgcn_wmma_f32_16x16x4_f32(false, av, false, b1,
                                                   (short)0, c1, false, false);
        c2 = __builtin_amdgcn_wmma_f32_16x16x4_f32(false, av, false, b2,
                                                   (short)0, c2, false, false);
    }

    // Epilogue: C layout — reg r: M = r + 8*half, N = lane&15
    const float na_b = na[bcol];
    const float s_ab = sab[a * BB + bcol];
#pragma unroll
    for (int r = 0; r < 8; ++r) {
        const int kidx = mtile * 16 + r + half * 8;
        if (kidx < KP1) {
            const float dotv = c1[r] - s_ab;
            float dsq = nw[a * KP1 + kidx] - 2.0f * c2[r] + na_b;
            dsq = dsq < 0.0f ? 0.0f : dsq;
            float den = sqrtf(dsq);
            den = den < 1e-12f ? 1e-12f : den;
            float o = dotv / den;
            o = o > 1.0f ? 1.0f : (o < -1.0f ? -1.0f : o);   // cosine bound
            eout[((size_t)a * BB + bcol) * KP1 + kidx] = expf(o * INV_T);
        }
    }
}

// ---------------------------------------------------------------------------
// Deterministic 2-stage sum for Z, then in-place scaling
// ---------------------------------------------------------------------------
__global__ __launch_bounds__(256) void reduce_partial_kernel(
    const float* __restrict__ e, float* __restrict__ part, int nelem)
{
    __shared__ float red[256];
    const int t = threadIdx.x;
    const size_t i = (size_t)blockIdx.x * 256 + t;
    red[t] = (i < (size_t)nelem) ? e[i] : 0.0f;
    __syncthreads();
    for (int s = 128; s > 0; s >>= 1) { if (t < s) red[t] += red[t + s]; __syncthreads(); }
    if (t == 0) part[blockIdx.x] = red[0];
}

__global__ __launch_bounds__(256) void finalize_scale_kernel(
    const float* __restrict__ part, int n, float* __restrict__ scale)
{
    __shared__ float red[256];
    const int t = threadIdx.x;
    float acc = 0.0f;
    for (int i = t; i < n; i += 256) acc += part[i];   // fixed order per thread
    red[t] = acc;
    __syncthreads();
    for (int s = 128; s > 0; s >>= 1) { if (t < s) red[t] += red[t + s]; __syncthreads(); }
    if (t == 0) {
        const float Z = red[0] / ((float)BB * (float)BB * (float)KP1) * (float)NN;
        scale[0] = 1.0f / Z;
    }
}

__global__ __launch_bounds__(256) void scale_kernel(
    float* __restrict__ e, const float* __restrict__ scale, int nelem)
{
    const size_t i = (size_t)blockIdx.x * 256 + threadIdx.x;
    if (i < (size_t)nelem) e[i] *= scale[0];
}

// ---------------------------------------------------------------------------
// Memory bank outputs: full copy (float4 grid-stride) + sequential row update
// (single block => deterministic last-wins for duplicate y, like JAX scatter)
// ---------------------------------------------------------------------------
__global__ __launch_bounds__(256) void copy_bank_kernel(
    const float4* __restrict__ src, float4* __restrict__ dst, int n4)
{
    int i = blockIdx.x * blockDim.x + threadIdx.x;
    const int stride = gridDim.x * blockDim.x;
    for (; i < n4; i += stride) dst[i] = src[i];
}

__global__ __launch_bounds__(128) void update_bank_kernel(
    const float* __restrict__ mem, const float* __restrict__ v,
    const int* __restrict__ y, float* __restrict__ outbank)
{
    __shared__ float red[128];
    const int t = threadIdx.x;
    for (int i = 0; i < BB; ++i) {
        const int row = y[i];
        const float p = mem[(size_t)row * DD + t] * MOM + v[i * DD + t] * (1.0f - MOM);
        red[t] = p * p;
        __syncthreads();
        for (int s = 64; s > 0; s >>= 1) { if (t < s) red[t] += red[t + s]; __syncthreads(); }
        const float nrm = sqrtf(red[0]);
        __syncthreads();
        outbank[(size_t)row * DD + t] = p / nrm;
    }
}

// ---------------------------------------------------------------------------
// Launch
// ---------------------------------------------------------------------------
extern "C" void kernel_launch(void* const* d_in, const int* in_sizes, int n_in,
                              void* d_out, int out_size, void* d_ws, size_t ws_size,
                              hipStream_t stream) {
    const float* v1   = (const float*)d_in[0];
    const float* v2   = (const float*)d_in[1];
    const float* mem1 = (const float*)d_in[2];
    const float* mem2 = (const float*)d_in[3];
    const int*   y    = (const int*)d_in[4];
    const int*   idx  = (const int*)d_in[5];

    float* out = (float*)d_out;

    // Workspace layout (float offsets)
    float* ws = (float*)d_ws;
    const size_t ANC   = (size_t)BB * DD;          // 8192
    const size_t WP    = (size_t)BB * KP1 * DD;    // 1,056,768
    const size_t ARN   = (size_t)BB * BB * DD;     // 524,288
    const size_t NSIDE = (size_t)BB * BB * KP1;    // 528,384 e-elements per side
    const int    NPART = (int)((NSIDE + 255) / 256);

    size_t o = 0;
    float* anc[2];  anc[0] = ws + o; o += ANC;  anc[1] = ws + o; o += ANC;
    float* na[2];   na[0]  = ws + o; o += BB;   na[1]  = ws + o; o += BB;
    float* wp[2];   wp[0]  = ws + o; o += WP;   wp[1]  = ws + o; o += WP;
    float* nw[2];   nw[0]  = ws + o; o += (size_t)BB * KP1;
                    nw[1]  = ws + o; o += (size_t)BB * KP1;
    float* arn[2];  arn[0] = ws + o; o += ARN;  arn[1] = ws + o; o += ARN;
    float* sab[2];  sab[0] = ws + o; o += (size_t)BB * BB;
                    sab[1] = ws + o; o += (size_t)BB * BB;
    float* part[2]; part[0] = ws + o; o += NPART; part[1] = ws + o; o += NPART;
    float* scale[2]; scale[0] = ws + o; o += 2; scale[1] = scale[0] + 1;

    // side 0: (v1 vs memory_v2) ; side 1: (v2 vs memory_v1)
    const float* vv[2]  = { v1, v2 };
    const float* mm[2]  = { mem2, mem1 };

    for (int s = 0; s < 2; ++s) {
        float* eout = out + (size_t)s * NSIDE;
        prep_anchor_kernel<<<BB, 128, 0, stream>>>(mm[s], y, anc[s], na[s]);
        prep_weight_kernel<<<BB * KP1, 128, 0, stream>>>(mm[s], idx, wp[s], nw[s]);
        prep_ar_kernel<<<BB * BB, 128, 0, stream>>>(vv[s], anc[s], arn[s], sab[s]);
        einsum_wmma_kernel<<<dim3(4, 9, BB), 32, 0, stream>>>(
            wp[s], arn[s], anc[s], sab[s], nw[s], na[s], eout);
        reduce_partial_kernel<<<NPART, 256, 0, stream>>>(eout, part[s], (int)NSIDE);
        finalize_scale_kernel<<<1, 256, 0, stream>>>(part[s], NPART, scale[s]);
        scale_kernel<<<NPART, 256, 0, stream>>>(eout, scale[s], (int)NSIDE);
    }

    // Momentum-updated memory banks
    float* out_m1 = out + 2 * NSIDE;
    float* out_m2 = out_m1 + (size_t)NN * DD;
    const int n4 = NN * DD / 4;
    copy_bank_kernel<<<2048, 256, 0, stream>>>((const float4*)mem1, (float4*)out_m1, n4);
    copy_bank_kernel<<<2048, 256, 0, stream>>>((const float4*)mem2, (float4*)out_m2, n4);
    update_bank_kernel<<<1, 128, 0, stream>>>(mem1, v1, y, out_m1);
    update_bank_kernel<<<1, 128, 0, stream>>>(mem2, v2, y, out_m2);
}